// CSA_47382079210108
// MI455X (gfx1250) — compile-verified
//
#include <hip/hip_runtime.h>
#include <math.h>

// Channel (cross-covariance) attention, fused LN + grouped 1x1 conv + attention.
// x: [1, 64, 64, 96, 96] fp32 -> treat as [C=64][N=589824].
// Pass1: LN + q,k -> per-block partial logits (64x64) via V_WMMA_F32_16X16X4_F32,
//        register double-buffered so HBM fill overlaps the WMMA phase.
// Pass2: parallel coalesced partial reduction + row softmax (deterministic order)
// Pass3: LN + v -> out = attn @ v + x via WMMA, staged + b128 nontemporal stores

#define CCH 64
#define TS 128
#define NPOS 589824           // 64*96*96
#define NP4  (NPOS / 4)
#define NTILES (NPOS / TS)    // 4608
#define NBLK1 256
#define XT_ST 132             // x tile stride: 16B-aligned rows, conflict-free frags
#define VT_ST 136             // v tile stride: conflict-free B-fragment reads
#define AT_ST 68              // attn stride: conflict-free A-fragment reads
#define RED_ST 68             // softmax reduction stride

typedef float v2f __attribute__((ext_vector_type(2)));
typedef float v4f __attribute__((ext_vector_type(4)));
typedef float v8f __attribute__((ext_vector_type(8)));

// ---------------- Pass 1: LN + q,k, accumulate partial logits ----------------
__global__ __launch_bounds__(256) void csa_pass1(
    const float* __restrict__ x, const float* __restrict__ gamma,
    const float* __restrict__ beta, const float* __restrict__ cw,
    const float* __restrict__ cbias, float* __restrict__ partials)
{
  __shared__ __align__(16) float xt[CCH * XT_ST];
  __shared__ __align__(16) float mu[TS];
  __shared__ __align__(16) float rs[TS];
  __shared__ float sg[CCH], sb[CCH], scw[3 * CCH], scb[3 * CCH];

  const int tid = threadIdx.x;
  if (tid < CCH)     { sg[tid] = gamma[tid]; sb[tid] = beta[tid]; }
  if (tid < 3 * CCH) { scw[tid] = cw[tid];   scb[tid] = cbias[tid]; }
  __syncthreads();

  const int wave = tid >> 5, lane = tid & 31;
  const int lh = lane >> 4, lm = lane & 15;
  // each wave owns 2 of the 16 (4x4) 16x16 logit tiles
  const int T0 = wave * 2, T1 = T0 + 1;
  const int cb0 = (T0 >> 2) * 16, kb0 = (T0 & 3) * 16;
  const int cb1 = (T1 >> 2) * 16, kb1 = (T1 & 3) * 16;
  const float wq0 = scw[3 * (cb0 + lm) + 0], bq0 = scb[3 * (cb0 + lm) + 0];
  const float wk0 = scw[3 * (kb0 + lm) + 1], bk0 = scb[3 * (kb0 + lm) + 1];
  const float wq1 = scw[3 * (cb1 + lm) + 0], bq1 = scb[3 * (cb1 + lm) + 0];
  const float wk1 = scw[3 * (kb1 + lm) + 1], bk1 = scb[3 * (kb1 + lm) + 1];

  const float* rowA0 = &xt[(cb0 + lm) * XT_ST + 2 * lh];
  const float* rowB0 = &xt[(kb0 + lm) * XT_ST + 2 * lh];
  const float* rowA1 = &xt[(cb1 + lm) * XT_ST + 2 * lh];
  const float* rowB1 = &xt[(kb1 + lm) * XT_ST + 2 * lh];

  const v4f* __restrict__ x4 = (const v4f*)x;
  const int c_ld  = tid >> 5;          // this thread's fixed (row, col4) slots
  const int p4_ld = tid & 31;

  v8f acc0 = {}; v8f acc1 = {};
  v4f buf[8];

  // prologue: load first tile into registers
  {
    const int t4 = blockIdx.x * (TS / 4);
    #pragma unroll
    for (int r = 0; r < 8; ++r)
      buf[r] = x4[(c_ld + r * 8) * NP4 + t4 + p4_ld];
  }

  for (int tile = blockIdx.x; tile < NTILES; tile += NBLK1) {
    // spill prefetched tile to LDS (b128 stores)
    #pragma unroll
    for (int r = 0; r < 8; ++r)
      *(v4f*)&xt[(c_ld + r * 8) * XT_ST + p4_ld * 4] = buf[r];
    __syncthreads();
    // per-position channel statistics
    if (tid < TS) {
      float s = 0.f, ss = 0.f;
      #pragma unroll 8
      for (int c = 0; c < CCH; ++c) {
        float v = xt[c * XT_ST + tid];
        s += v; ss += v * v;
      }
      float m = s * (1.0f / CCH);
      mu[tid] = m;
      rs[tid] = rsqrtf(ss * (1.0f / CCH) - m * m + 1e-5f);
    }
    __syncthreads();
    // normalize in place: y = (x - mu) * rstd * gamma + beta  (vectorized)
    #pragma unroll
    for (int r = 0; r < 8; ++r) {
      int c = c_ld + r * 8;
      v4f xv  = *(v4f*)&xt[c * XT_ST + p4_ld * 4];
      v4f muv = *(v4f*)&mu[p4_ld * 4];
      v4f rsv = *(v4f*)&rs[p4_ld * 4];
      *(v4f*)&xt[c * XT_ST + p4_ld * 4] = (xv - muv) * rsv * sg[c] + sb[c];
    }
    __syncthreads();
    // prefetch NEXT tile into registers; loads stay in flight across the
    // whole WMMA phase (no dependence until next iteration's LDS spill)
    const int next = tile + NBLK1;
    if (next < NTILES) {
      const int t4n = next * (TS / 4);
      #pragma unroll
      for (int r = 0; r < 8; ++r)
        buf[r] = x4[(c_ld + r * 8) * NP4 + t4n + p4_ld];
    }
    // logits += q-tile (16xK) x k-tile (Kx16), K in steps of 4 positions
    for (int p0 = 0; p0 < TS; p0 += 4) {
      v2f a, b;
      a.x = rowA0[p0] * wq0 + bq0;  a.y = rowA0[p0 + 1] * wq0 + bq0;
      b.x = rowB0[p0] * wk0 + bk0;  b.y = rowB0[p0 + 1] * wk0 + bk0;
      acc0 = __builtin_amdgcn_wmma_f32_16x16x4_f32(false, a, false, b,
                                                   (short)0, acc0, false, false);
      a.x = rowA1[p0] * wq1 + bq1;  a.y = rowA1[p0 + 1] * wq1 + bq1;
      b.x = rowB1[p0] * wk1 + bk1;  b.y = rowB1[p0 + 1] * wk1 + bk1;
      acc1 = __builtin_amdgcn_wmma_f32_16x16x4_f32(false, a, false, b,
                                                   (short)0, acc1, false, false);
    }
    __syncthreads();
  }

  // write this block's partial 64x64 logits (deterministic, no atomics)
  float* pt = partials + blockIdx.x * (CCH * CCH);
  union { v8f v; float f[8]; } u0, u1;
  u0.v = acc0; u1.v = acc1;
  #pragma unroll
  for (int i = 0; i < 8; ++i) {
    pt[(cb0 + i + 8 * lh) * CCH + kb0 + lm] = u0.f[i];
    pt[(cb1 + i + 8 * lh) * CCH + kb1 + lm] = u1.f[i];
  }
}

// ------- Pass 2: coalesced parallel reduction of partials + row softmax -------
__global__ __launch_bounds__(1024) void csa_softmax(
    const float* __restrict__ partials, float* __restrict__ attn)
{
  __shared__ __align__(16) float red[CCH * RED_ST];
  const int tid = threadIdx.x;  // 1024 threads; one float4 of the 64x64 each
  const v4f* __restrict__ p4 = (const v4f*)partials;
  v4f s = {};
  for (int b = 0; b < NBLK1; ++b)
    s += p4[b * (CCH * CCH / 4) + tid];       // fully coalesced, fixed order
  const int e = tid * 4;
  const int row = e >> 6, col = e & 63;
  *(v4f*)&red[row * RED_ST + col] = s;
  __syncthreads();

  if (tid < CCH) {
    const float scale = 0.125f;  // C^-0.5
    const float* r = &red[tid * RED_ST];
    float m = -INFINITY;
    #pragma unroll 8
    for (int k = 0; k < CCH; ++k) m = fmaxf(m, r[k] * scale);
    float rowv[CCH];
    float sum = 0.f;
    #pragma unroll 8
    for (int k = 0; k < CCH; ++k) { rowv[k] = expf(r[k] * scale - m); sum += rowv[k]; }
    const float inv = 1.0f / sum;
    #pragma unroll 8
    for (int k = 0; k < CCH; ++k) attn[tid * CCH + k] = rowv[k] * inv;
  }
}

// ---------------- Pass 3: LN + v, out = attn @ v + x ----------------
__global__ __launch_bounds__(256) void csa_pass3(
    const float* __restrict__ x, const float* __restrict__ gamma,
    const float* __restrict__ beta, const float* __restrict__ cw,
    const float* __restrict__ cbias, const float* __restrict__ attn,
    float* __restrict__ out)
{
  __shared__ __align__(16) float xt[CCH * XT_ST];
  __shared__ __align__(16) float vt[CCH * VT_ST];  // v tile, then reused for D tiles
  __shared__ __align__(16) float at[CCH * AT_ST];
  __shared__ __align__(16) float mu[TS];
  __shared__ __align__(16) float rs[TS];
  __shared__ float sg[CCH], sb[CCH], swv[CCH], sbv[CCH];

  const int tid = threadIdx.x;
  if (tid < CCH) {
    sg[tid] = gamma[tid]; sb[tid] = beta[tid];
    swv[tid] = cw[3 * tid + 2]; sbv[tid] = cbias[3 * tid + 2];
  }
  for (int idx = tid; idx < CCH * CCH; idx += 256)
    at[(idx >> 6) * AT_ST + (idx & 63)] = attn[idx];

  const int n0 = blockIdx.x * TS;
  const int t4 = blockIdx.x * (TS / 4);
  const v4f* __restrict__ x4 = (const v4f*)x;
  #pragma unroll
  for (int r = 0; r < 8; ++r) {
    int idx = r * 256 + tid;
    int c = idx >> 5, p4 = idx & 31;
    *(v4f*)&xt[c * XT_ST + p4 * 4] = x4[c * NP4 + t4 + p4];
  }
  __syncthreads();
  if (tid < TS) {
    float s = 0.f, ss = 0.f;
    #pragma unroll 8
    for (int c = 0; c < CCH; ++c) {
      float v = xt[c * XT_ST + tid];
      s += v; ss += v * v;
    }
    float m = s * (1.0f / CCH);
    mu[tid] = m;
    rs[tid] = rsqrtf(ss * (1.0f / CCH) - m * m + 1e-5f);
  }
  __syncthreads();
  // v = LN(x) * wv + bv  (x kept intact for residual)
  #pragma unroll
  for (int r = 0; r < 8; ++r) {
    int idx = r * 256 + tid;
    int c = idx >> 5, p4 = idx & 31;
    v4f xv  = *(v4f*)&xt[c * XT_ST + p4 * 4];
    v4f muv = *(v4f*)&mu[p4 * 4];
    v4f rsv = *(v4f*)&rs[p4 * 4];
    v4f y = (xv - muv) * rsv * sg[c] + sb[c];
    *(v4f*)&vt[c * VT_ST + p4 * 4] = y * swv[c] + sbv[c];
  }
  __syncthreads();

  const int wave = tid >> 5, lane = tid & 31;
  const int lh = lane >> 4, lm = lane & 15;
  // 32 output tiles (4 row-groups x 8 col-groups); 4 per wave
  v8f acc[4];
  #pragma unroll
  for (int t = 0; t < 4; ++t) {
    const int T = wave * 4 + t;
    const int mb = (T >> 3) * 16, pb = (T & 7) * 16;
    v8f a8 = {};
    const float* arow = &at[(mb + lm) * AT_ST + 2 * lh];
    const float* bcol = &vt[(2 * lh) * VT_ST + pb + lm];
    for (int kc = 0; kc < CCH; kc += 4) {
      v2f a, b;
      a.x = arow[kc];          a.y = arow[kc + 1];
      b.x = bcol[kc * VT_ST];  b.y = bcol[(kc + 1) * VT_ST];
      a8 = __builtin_amdgcn_wmma_f32_16x16x4_f32(false, a, false, b,
                                                 (short)0, a8, false, false);
    }
    acc[t] = a8;
  }
  __syncthreads();   // all waves done reading vt -> reuse as D-tile staging
  #pragma unroll
  for (int t = 0; t < 4; ++t) {
    const int T = wave * 4 + t;
    const int mb = (T >> 3) * 16, pb = (T & 7) * 16;
    union { v8f v; float f[8]; } u; u.v = acc[t];
    #pragma unroll
    for (int i = 0; i < 8; ++i)
      vt[(mb + i + 8 * lh) * VT_ST + pb + lm] = u.f[i];
  }
  __syncthreads();
  // coalesced b128 nontemporal stores: out = D + x (keep x resident in L2)
  #pragma unroll
  for (int r = 0; r < 8; ++r) {
    int idx = r * 256 + tid;
    int c = idx >> 5, p4 = idx & 31;
    v4f o = *(v4f*)&vt[c * VT_ST + p4 * 4];
    v4f xv = *(v4f*)&xt[c * XT_ST + p4 * 4];
    v4f sum = o + xv;
    __builtin_nontemporal_store(sum, (v4f*)&out[c * NPOS + n0 + p4 * 4]);
  }
}

extern "C" void kernel_launch(void* const* d_in, const int* in_sizes, int n_in,
                              void* d_out, int out_size, void* d_ws, size_t ws_size,
                              hipStream_t stream) {
  (void)in_sizes; (void)n_in; (void)out_size; (void)ws_size;
  const float* x     = (const float*)d_in[0];
  const float* gamma = (const float*)d_in[1];
  const float* beta  = (const float*)d_in[2];
  const float* cw    = (const float*)d_in[3];
  const float* cbias = (const float*)d_in[4];
  float* out = (float*)d_out;

  float* partials = (float*)d_ws;                          // NBLK1 * 64 * 64 floats
  float* attn     = partials + (size_t)NBLK1 * CCH * CCH;  // 64 * 64 floats

  csa_pass1<<<NBLK1, 256, 0, stream>>>(x, gamma, beta, cw, cbias, partials);
  csa_softmax<<<1, 1024, 0, stream>>>(partials, attn);
  csa_pass3<<<NTILES, 256, 0, stream>>>(x, gamma, beta, cw, cbias, attn, out);
}